// ModelNew_73315091743759
// MI455X (gfx1250) — compile-verified
//
#include <hip/hip_runtime.h>
#include <math.h>

typedef __attribute__((ext_vector_type(2))) float v2f;
typedef __attribute__((ext_vector_type(8))) float v8f;

#define TILE 128
#define BK 16
#define A_STRIDE 20    // 16 + pad: consecutive rows land in distinct banks
#define BP_STRIDE 288  // 128 cols * 2 (k-pair) + 32 pad: lane halves on disjoint bank groups
#define NPAIR (BK / 2)

__global__ __launch_bounds__(256) void triu_gemm_wmma_f32(
    const float* __restrict__ A, const float* __restrict__ B,
    float* __restrict__ C, int N) {
  const int bj = blockIdx.x;  // column tile index
  const int bi = blockIdx.y;  // row tile index
  const int row0 = bi * TILE;
  const int col0 = bj * TILE;
  const int t = threadIdx.x;

  if (bi > bj) {
    // Strictly-lower-triangular block: output is identically zero.
    float4 z = make_float4(0.f, 0.f, 0.f, 0.f);
#pragma unroll
    for (int r = 0; r < 16; ++r) {
      int linear = t + 256 * r;     // 4096 float4 = 128*128 floats
      int row = linear >> 5;        // 0..127
      int c4 = (linear & 31) << 2;  // 0..124
      *(float4*)(C + (size_t)(row0 + row) * N + col0 + c4) = z;
    }
    return;
  }

  __shared__ alignas(16) float As[2][TILE * A_STRIDE];
  __shared__ alignas(16) float Bs[2][NPAIR * BP_STRIDE];

  const int lane = t & 31;
  const int wave = t >> 5;
  const int wm = wave >> 2;  // 0..1 : wave row (64 rows each)
  const int wn = wave & 3;   // 0..3 : wave col (32 cols each)
  const int l16 = lane & 15;
  const int lh = lane >> 4;  // 0/1 : lane half

  // Per-thread tile-load coordinates (invariant across K tiles).
  const int arow0 = t >> 2;        // A: rows 0..63 (r=1 adds 64)
  const int ac = (t & 3) << 2;     // A: k-subcol 0,4,8,12
  const int pair0 = t >> 6;        // B: k-pairs 0..3 (r=1 adds 4)
  const int colp = t & 63;         // B: column pair 0..63

  // Staging registers for the software pipeline (global -> regs -> LDS).
  float4 ra[2];
  float2 rbu[2], rbv[2];

  auto g_load = [&](int kt) {
#pragma unroll
    for (int r = 0; r < 2; ++r) {
      const int arow = arow0 + 64 * r;
      ra[r] = *(const float4*)(A + (size_t)(row0 + arow) * N + kt + ac);
      const int pr = pair0 + 4 * r;
      const int k0 = kt + pr * 2;
      const float* bp = B + (size_t)k0 * N + col0 + colp * 2;
      rbu[r] = *(const float2*)bp;        // row k0
      rbv[r] = *(const float2*)(bp + N);  // row k0+1
    }
  };

  auto l_store = [&](int p, int kt) {
#pragma unroll
    for (int r = 0; r < 2; ++r) {
      // A tile with triu(A) mask: keep k >= i
      const int arow = arow0 + 64 * r;
      const int i = row0 + arow;
      const int k = kt + ac;
      float4 m;
      m.x = (k + 0 >= i) ? ra[r].x : 0.f;
      m.y = (k + 1 >= i) ? ra[r].y : 0.f;
      m.z = (k + 2 >= i) ? ra[r].z : 0.f;
      m.w = (k + 3 >= i) ? ra[r].w : 0.f;
      *(float4*)(&As[p][arow * A_STRIDE + ac]) = m;

      // B tile, k-pair packed, with triu(B) mask: keep k <= j
      const int pr = pair0 + 4 * r;
      const int k0 = kt + pr * 2;
      const int c = col0 + colp * 2;
      float4 pk;
      pk.x = (k0 + 0 <= c + 0) ? rbu[r].x : 0.f;  // B[k0  ][c  ]
      pk.y = (k0 + 1 <= c + 0) ? rbv[r].x : 0.f;  // B[k0+1][c  ]
      pk.z = (k0 + 0 <= c + 1) ? rbu[r].y : 0.f;  // B[k0  ][c+1]
      pk.w = (k0 + 1 <= c + 1) ? rbv[r].y : 0.f;  // B[k0+1][c+1]
      *(float4*)(&Bs[p][pr * BP_STRIDE + colp * 4]) = pk;
    }
  };

  v8f acc[4][2];
#pragma unroll
  for (int mi = 0; mi < 4; ++mi)
#pragma unroll
    for (int ni = 0; ni < 2; ++ni) {
      v8f z = {};
      acc[mi][ni] = z;
    }

  // k must satisfy row0 <= i <= k <= j <= col0+TILE-1
  const int kBeg = row0;
  const int kEnd = col0 + TILE;

  // Pipeline prologue
  g_load(kBeg);
  l_store(0, kBeg);
  __syncthreads();

  int p = 0;
  for (int kt = kBeg; kt < kEnd; kt += BK) {
    const bool hasNext = (kt + BK) < kEnd;
    if (hasNext) g_load(kt + BK);  // global loads in flight during compute

#pragma unroll
    for (int ks = 0; ks < BK; ks += 4) {
      // A fragment 16x4: lanes 0-15 K={ks,ks+1}, lanes 16-31 K={ks+2,ks+3}
      v2f afrag[4];
#pragma unroll
      for (int mi = 0; mi < 4; ++mi) {
        const int arow = wm * 64 + mi * 16 + l16;
        afrag[mi] = *(const v2f*)(&As[p][arow * A_STRIDE + ks + 2 * lh]);
      }
      // B fragment 4x16: single contiguous pair load per fragment (no repack)
      v2f bfrag[2];
#pragma unroll
      for (int ni = 0; ni < 2; ++ni) {
        const int bcol = wn * 32 + ni * 16 + l16;
        bfrag[ni] =
            *(const v2f*)(&Bs[p][((ks >> 1) + lh) * BP_STRIDE + bcol * 2]);
      }
#pragma unroll
      for (int mi = 0; mi < 4; ++mi)
#pragma unroll
        for (int ni = 0; ni < 2; ++ni)
          acc[mi][ni] = __builtin_amdgcn_wmma_f32_16x16x4_f32(
              false, afrag[mi], false, bfrag[ni], (short)0, acc[mi][ni],
              false, false);
    }

    if (hasNext) {
      l_store(p ^ 1, kt + BK);
      __syncthreads();  // single barrier per K tile (double-buffered)
      p ^= 1;
    }
  }

  // ---- Store with triu(C) output mask (zeros below diagonal) ----
#pragma unroll
  for (int mi = 0; mi < 4; ++mi) {
#pragma unroll
    for (int ni = 0; ni < 2; ++ni) {
      const int col = col0 + wn * 32 + ni * 16 + l16;
#pragma unroll
      for (int v = 0; v < 8; ++v) {
        const int row = row0 + wm * 64 + mi * 16 + v + 8 * lh;
        const float val = (row <= col) ? acc[mi][ni][v] : 0.f;
        C[(size_t)row * N + col] = val;
      }
    }
  }
}

extern "C" void kernel_launch(void* const* d_in, const int* in_sizes, int n_in,
                              void* d_out, int out_size, void* d_ws, size_t ws_size,
                              hipStream_t stream) {
  const float* A = (const float*)d_in[0];
  const float* B = (const float*)d_in[1];
  float* C = (float*)d_out;
  int N = (int)llround(sqrt((double)in_sizes[0]));  // 8192
  dim3 grid(N / TILE, N / TILE);
  triu_gemm_wmma_f32<<<grid, 256, 0, stream>>>(A, B, C, N);
}